// TimeDistributedHierarchicalSoftmax_38465727103178
// MI455X (gfx1250) — compile-verified
//
#include <hip/hip_runtime.h>
#include <math.h>

// ---------------------------------------------------------------------------
// TimeDistributedHierarchicalSoftmax on MI455X (gfx1250, wave32)
//
// N=2048 samples, D=1024, 224 classes x 224 per-class.
// Strategy: group samples by class so each 918KB expert matrix is read once
// (~205MB total ~ 9us at 23.3 TB/s), then do fp32 WMMA (16x16x4) GEMMs.
// ---------------------------------------------------------------------------

typedef float v2f __attribute__((ext_vector_type(2)));
typedef float v8f __attribute__((ext_vector_type(8)));

#define N_SAMPLES 2048
#define DIM 1024
#define NCLS 224
#define PCLS 224
#define NTILES 14          // 224 / 16

__device__ __forceinline__ v8f wmma_f32(v2f a, v2f b, v8f c) {
  // D = A(16x4 f32) * B(4x16 f32) + C(16x16 f32)
  return __builtin_amdgcn_wmma_f32_16x16x4_f32(
      /*neg_a=*/false, a, /*neg_b=*/false, b,
      /*c_mod=*/(short)0, c, /*reuse_a=*/false, /*reuse_b=*/false);
}

// ---------------------------------------------------------------------------
// Kernel 1: top logits = x @ top_weights + top_bias  (2048x1024 * 1024x224)
// 8 waves per block; each wave owns a 16-row sample tile and all 14 N-tiles.
// ---------------------------------------------------------------------------
__global__ void __launch_bounds__(256)
top_gemm_kernel(const float* __restrict__ x, const float* __restrict__ wt,
                const float* __restrict__ bt, float* __restrict__ logits)
{
  const int lane  = threadIdx.x & 31;
  const int wave  = threadIdx.x >> 5;
  const int laneM = lane & 15;
  const int kOff  = (lane >> 4) << 1;   // 0 (lanes 0-15) or 2 (lanes 16-31)
  const int m0    = (blockIdx.x * 8 + wave) * 16;

  v8f acc[NTILES];
#pragma unroll
  for (int i = 0; i < NTILES; ++i) acc[i] = {};

  const float* rowA = x + (size_t)(m0 + laneM) * DIM;

  for (int k = 0; k < DIM; k += 4) {
    v2f a;
    a.x = rowA[k + kOff];
    a.y = rowA[k + kOff + 1];
#pragma unroll
    for (int nt = 0; nt < NTILES; ++nt) {
      const int col = nt * 16 + laneM;
      v2f b;
      b.x = wt[(size_t)(k + kOff) * NCLS + col];
      b.y = wt[(size_t)(k + kOff + 1) * NCLS + col];
      acc[nt] = wmma_f32(a, b, acc[nt]);
    }
  }

  const int rBase = (lane >> 4) << 3;   // 0 or 8
#pragma unroll
  for (int nt = 0; nt < NTILES; ++nt) {
    const int col  = nt * 16 + laneM;
    const float bv = bt[col];
#pragma unroll
    for (int v = 0; v < 8; ++v) {
      logits[(size_t)(m0 + rBase + v) * NCLS + col] = acc[nt][v] + bv;
    }
  }
}

// ---------------------------------------------------------------------------
// Kernel 2: per-sample top softmax, pick p_top[n, cls[n]].  One wave / sample.
// ---------------------------------------------------------------------------
__global__ void __launch_bounds__(256)
top_pick_kernel(const float* __restrict__ logits, const int* __restrict__ tgt,
                float* __restrict__ ptop)
{
  const int lane = threadIdx.x & 31;
  const int wave = threadIdx.x >> 5;
  const int n = blockIdx.x * 8 + wave;
  const float* row = logits + (size_t)n * NCLS;

  float vals[7];
  float mx = -3.4e38f;
#pragma unroll
  for (int i = 0; i < 7; ++i) { vals[i] = row[lane + 32 * i]; mx = fmaxf(mx, vals[i]); }
#pragma unroll
  for (int m = 16; m >= 1; m >>= 1) mx = fmaxf(mx, __shfl_xor(mx, m, 32));
  float sm = 0.f;
#pragma unroll
  for (int i = 0; i < 7; ++i) sm += __expf(vals[i] - mx);
#pragma unroll
  for (int m = 16; m >= 1; m >>= 1) sm += __shfl_xor(sm, m, 32);

  if (lane == 0) {
    const int cls = tgt[n] / PCLS;
    ptop[n] = __expf(row[cls] - mx) / sm;
  }
}

// ---------------------------------------------------------------------------
// Class-grouping helpers: zero, histogram, scan, scatter.
// ---------------------------------------------------------------------------
__global__ void zero_kernel(int* counts, int* cursor) {
  const int i = threadIdx.x;
  if (i < NCLS) { counts[i] = 0; cursor[i] = 0; }
}

__global__ void hist_kernel(const int* __restrict__ tgt, int* counts) {
  const int i = blockIdx.x * blockDim.x + threadIdx.x;
  if (i < N_SAMPLES) atomicAdd(&counts[tgt[i] / PCLS], 1);
}

__global__ void scan_kernel(const int* __restrict__ counts, int* offsets) {
  if (threadIdx.x == 0) {
    int run = 0;
    for (int c = 0; c < NCLS; ++c) { offsets[c] = run; run += counts[c]; }
    offsets[NCLS] = run;
  }
}

__global__ void scatter_kernel(const int* __restrict__ tgt,
                               const int* __restrict__ offsets,
                               int* cursor, int* order) {
  const int i = blockIdx.x * blockDim.x + threadIdx.x;
  if (i < N_SAMPLES) {
    const int cls = tgt[i] / PCLS;
    const int pos = offsets[cls] + atomicAdd(&cursor[cls], 1);
    order[pos] = i;
  }
}

// ---------------------------------------------------------------------------
// Kernel 3: per-class bottom GEMM + softmax + final product.
// One block (8 waves) per class. x rows staged in LDS in 256-wide K chunks.
// ---------------------------------------------------------------------------
__global__ void __launch_bounds__(256)
bottom_kernel(const float* __restrict__ x, const int* __restrict__ tgt,
              const float* __restrict__ bw, const float* __restrict__ bb,
              const float* __restrict__ ptop, const int* __restrict__ offsets,
              const int* __restrict__ order, float* __restrict__ out)
{
  __shared__ float xs[16][256];      // 16 KB: K-chunk of gathered x rows
  __shared__ float lg[16][NCLS];     // 14 KB: bottom logits tile
  __shared__ int   sidx[16];

  const int c     = blockIdx.x;
  const int start = offsets[c];
  const int cnt   = offsets[c + 1] - start;
  if (cnt == 0) return;              // block-uniform

  const int tid   = threadIdx.x;
  const int lane  = tid & 31;
  const int wave  = tid >> 5;
  const int laneM = lane & 15;
  const int kOff  = (lane >> 4) << 1;
  const int rBase = (lane >> 4) << 3;

  const float* Wc = bw + (size_t)c * DIM * PCLS;

  const int nt0   = wave;            // < 8  (always valid)
  const int nt1   = wave + 8;        // valid for waves 0..5
  const bool has1 = (nt1 < NTILES);  // wave-uniform
  const int col0  = nt0 * 16 + laneM;
  const int col1  = has1 ? nt1 * 16 + laneM : 0;

  const int r  = tid >> 4;           // row this thread stages (0..15)
  const int cw = tid & 15;           // 16 loaders per row

  for (int t0 = 0; t0 < cnt; t0 += 16) {
    __syncthreads();                 // previous tile fully consumed
    if (tid < 16) sidx[tid] = (t0 + tid < cnt) ? order[start + t0 + tid] : -1;
    __syncthreads();

    v8f acc0 = {}, acc1 = {};

    for (int kc = 0; kc < DIM; kc += 256) {
      // stage x rows chunk into LDS (16 floats per thread, float4)
      {
        const int s = sidx[r];
        float4* dst = (float4*)&xs[r][cw * 16];
        if (s >= 0) {
          const float4* src = (const float4*)(x + (size_t)s * DIM + kc + cw * 16);
#pragma unroll
          for (int j = 0; j < 4; ++j) dst[j] = src[j];
        } else {
          const float4 z = make_float4(0.f, 0.f, 0.f, 0.f);
#pragma unroll
          for (int j = 0; j < 4; ++j) dst[j] = z;
        }
      }
      __syncthreads();

      const float* WcK = Wc + (size_t)kc * PCLS;
      for (int k = 0; k < 256; k += 4) {
        v2f a;
        a.x = xs[laneM][k + kOff];
        a.y = xs[laneM][k + kOff + 1];
        v2f b0;
        b0.x = WcK[(size_t)(k + kOff) * PCLS + col0];
        b0.y = WcK[(size_t)(k + kOff + 1) * PCLS + col0];
        acc0 = wmma_f32(a, b0, acc0);
        if (has1) {
          v2f b1;
          b1.x = WcK[(size_t)(k + kOff) * PCLS + col1];
          b1.y = WcK[(size_t)(k + kOff + 1) * PCLS + col1];
          acc1 = wmma_f32(a, b1, acc1);
        }
      }
      __syncthreads();
    }

    // logits tile (+bias) -> LDS
    {
      const float bias0 = bb[(size_t)c * PCLS + col0];
#pragma unroll
      for (int v = 0; v < 8; ++v) lg[rBase + v][col0] = acc0[v] + bias0;
      if (has1) {
        const float bias1 = bb[(size_t)c * PCLS + col1];
#pragma unroll
        for (int v = 0; v < 8; ++v) lg[rBase + v][col1] = acc1[v] + bias1;
      }
    }
    __syncthreads();

    // per-row softmax + final output: 16 threads per sample row
    {
      const int g  = tid >> 4;
      const int ln = tid & 15;
      const int s  = sidx[g];
      float mx = -3.4e38f;
      for (int j = ln; j < NCLS; j += 16) mx = fmaxf(mx, lg[g][j]);
#pragma unroll
      for (int m = 8; m >= 1; m >>= 1) mx = fmaxf(mx, __shfl_xor(mx, m, 32));
      float sm = 0.f;
      for (int j = ln; j < NCLS; j += 16) sm += __expf(lg[g][j] - mx);
#pragma unroll
      for (int m = 8; m >= 1; m >>= 1) sm += __shfl_xor(sm, m, 32);
      if (ln == 0 && s >= 0) {
        const int within = tgt[s] % PCLS;
        out[s] = ptop[s] * __expf(lg[g][within] - mx) / sm;
      }
    }
  }
}

// ---------------------------------------------------------------------------
// Launch
// ---------------------------------------------------------------------------
extern "C" void kernel_launch(void* const* d_in, const int* in_sizes, int n_in,
                              void* d_out, int out_size, void* d_ws, size_t ws_size,
                              hipStream_t stream) {
  const float* x   = (const float*)d_in[0];   // (B,T,D) f32
  const int*   tgt = (const int*)  d_in[1];   // (B,T)   i32
  const float* wt  = (const float*)d_in[2];   // (D,NCLS)
  const float* bt  = (const float*)d_in[3];   // (NCLS,)
  const float* bw  = (const float*)d_in[4];   // (NCLS,D,PCLS)
  const float* bb  = (const float*)d_in[5];   // (NCLS,PCLS)
  float* out = (float*)d_out;                 // (B,T,1) f32

  // workspace layout
  float* top_logits = (float*)d_ws;                        // 2048*224 f32
  float* ptop       = top_logits + (size_t)N_SAMPLES * NCLS;  // 2048 f32
  int*   counts     = (int*)(ptop + N_SAMPLES);            // 224
  int*   offsets    = counts + NCLS;                       // 225
  int*   cursor     = offsets + (NCLS + 1);                // 224
  int*   order      = cursor + NCLS;                       // 2048

  zero_kernel<<<1, 256, 0, stream>>>(counts, cursor);
  top_gemm_kernel<<<N_SAMPLES / (16 * 8), 256, 0, stream>>>(x, wt, bt, top_logits);
  top_pick_kernel<<<N_SAMPLES / 8, 256, 0, stream>>>(top_logits, tgt, ptop);
  hist_kernel<<<N_SAMPLES / 256, 256, 0, stream>>>(tgt, counts);
  scan_kernel<<<1, 32, 0, stream>>>(counts, offsets);
  scatter_kernel<<<N_SAMPLES / 256, 256, 0, stream>>>(tgt, offsets, cursor, order);
  bottom_kernel<<<NCLS, 256, 0, stream>>>(x, tgt, bw, bb, ptop, offsets, order, out);
}